// GuidedFilter_29025388986986
// MI455X (gfx1250) — compile-verified
//
#include <hip/hip_runtime.h>
#include <stdint.h>

// ---------------------------------------------------------------------------
// Guided filter (r=1, eps=0.01), fused single pass for MI455X (gfx1250).
// Per-wave 16x16 tile; stage-1 box filters via LDS stencil; stage-2 box of
// A/b via V_WMMA_F32_16X16X4_F32 against tridiagonal 1.0 matrices (exact).
// Tile loads via global_load_async_to_lds_b64 (ASYNCcnt) with zero-fill halo.
// ---------------------------------------------------------------------------

#define IMG_H 768
#define IMG_W 768
#define TILES_X (IMG_W / 16)
#define TILES_PER_PLANE ((IMG_H / 16) * (IMG_W / 16))
#define WPB 4  // waves per block (wave32)

typedef __attribute__((ext_vector_type(2))) float v2f;
typedef __attribute__((ext_vector_type(8))) float v8f;
typedef int v2i_t __attribute__((vector_size(2 * sizeof(int))));

// per-wave LDS layout (float indices)
#define G_OFF   0      // 20x20 guide (stride 20)
#define S_OFF   400    // 20x20 src
#define HG_OFF  800    // 20x18 horizontal sums of g
#define HS_OFF  1160   // 20x18 of s
#define HGS_OFF 1520   // 20x18 of g*s
#define HGG_OFF 1880   // 20x18 of g*g
#define AM_OFF  2240   // 18x18 A map
#define BM_OFF  2564   // 18x18 b map
#define HT_OFF  2888   // 16x16 transposed intermediate, stride 17 (272)
#define WAVE_FLOATS 3168

// ---- async global->LDS copy (8 bytes per lane) ----------------------------
#if __has_builtin(__builtin_amdgcn_global_load_async_to_lds_b64)
#define ASYNC_BUILTIN 1
#else
#define ASYNC_BUILTIN 0
#endif

__device__ __forceinline__ void async_b64(const float* g, float* l) {
#if ASYNC_BUILTIN
  float* gnc = const_cast<float*>(g);
  __attribute__((address_space(1))) v2i_t* ga =
      (__attribute__((address_space(1))) v2i_t*)gnc;
  __attribute__((address_space(3))) v2i_t* la =
      (__attribute__((address_space(3))) v2i_t*)l;
  __builtin_amdgcn_global_load_async_to_lds_b64(ga, la, 0, 0);
#else
  uint32_t lds_addr =
      (uint32_t)(uintptr_t)(__attribute__((address_space(3))) void*)l;
  asm volatile("global_load_async_to_lds_b64 %0, %1, off"
               :: "v"(lds_addr), "v"(g)
               : "memory");
#endif
}

__device__ __forceinline__ void wait_async0() {
#if __has_builtin(__builtin_amdgcn_s_wait_asynccnt)
  __builtin_amdgcn_s_wait_asynccnt(0);
  asm volatile("" ::: "memory");
#else
  asm volatile("s_wait_asynccnt 0x0" ::: "memory");
#endif
}

// ---- f32 WMMA helper ------------------------------------------------------
__device__ __forceinline__ v8f wmma4(v2f a, v2f b, v8f c) {
  // D(16x16,f32) = A(16x4,f32) x B(4x16,f32) + C
  return __builtin_amdgcn_wmma_f32_16x16x4_f32(false, a, false, b, (short)0, c,
                                               false, false);
}

// 3x3 box sum of an 18x18 LDS map (stride 18) -> 16x16 result in C/D layout.
// S = Tv * (X * Th) with tridiagonal Tv/Th over the 16x16 core, plus VALU
// fixes for the halo column/row contributions the core matmul cannot see.
__device__ __forceinline__ v8f box3x3_wmma(float* lw, int Moff, int l, int h) {
  // ---- horizontal: Hcore = X * Th, X[m][k] = M[(m+1)*18 + (k+1)]
  v8f acc = {0.f, 0.f, 0.f, 0.f, 0.f, 0.f, 0.f, 0.f};
#pragma unroll
  for (int kk = 0; kk < 4; ++kk) {
    const int K0 = 4 * kk + 2 * h;  // A layout: vgpr0 holds K0, K0+2 halves
    v2f a, b;
    a.x = lw[Moff + (l + 1) * 18 + K0 + 1];
    a.y = lw[Moff + (l + 1) * 18 + K0 + 2];
    const int d0 = K0 - l, d1 = K0 + 1 - l;
    b.x = (d0 >= -1 && d0 <= 1) ? 1.0f : 0.0f;  // Th[K0][l]
    b.y = (d1 >= -1 && d1 <= 1) ? 1.0f : 0.0f;  // Th[K0+1][l]
    acc = wmma4(a, b, acc);
  }
  // column fixes: col 0 misses M[row][0], col 15 misses M[row][17]
  {
    const float f0 = (l == 0) ? 1.0f : 0.0f;
    const float f15 = (l == 15) ? 1.0f : 0.0f;
#pragma unroll
    for (int v = 0; v < 8; ++v) {
      const int row = 8 * h + v + 1;  // C/D layout: vgpr v -> rows v, v+8
      acc[v] += f0 * lw[Moff + row * 18 + 0] + f15 * lw[Moff + row * 18 + 17];
    }
  }
  // store transposed so it can be re-read in B layout
#pragma unroll
  for (int v = 0; v < 8; ++v) lw[HT_OFF + l * 17 + 8 * h + v] = acc[v];

  // ---- vertical: S = Tv * Y, Y[m][n] = HT[n][m]
  v8f acc2 = {0.f, 0.f, 0.f, 0.f, 0.f, 0.f, 0.f, 0.f};
#pragma unroll
  for (int kk = 0; kk < 4; ++kk) {
    const int K0 = 4 * kk + 2 * h;
    v2f a, b;
    const int d0 = l - K0, d1 = l - K0 - 1;
    a.x = (d0 >= -1 && d0 <= 1) ? 1.0f : 0.0f;  // Tv[l][K0]
    a.y = (d1 >= -1 && d1 <= 1) ? 1.0f : 0.0f;  // Tv[l][K0+1]
    b.x = lw[HT_OFF + l * 17 + K0];
    b.y = lw[HT_OFF + l * 17 + K0 + 1];
    acc2 = wmma4(a, b, acc2);
  }
  // row fixes: out row 0 misses Hh[0], out row 15 misses Hh[17]
  const float top = lw[Moff + l] + lw[Moff + l + 1] + lw[Moff + l + 2];
  const float bot = lw[Moff + 17 * 18 + l] + lw[Moff + 17 * 18 + l + 1] +
                    lw[Moff + 17 * 18 + l + 2];
  if (h == 0) acc2[0] += top;   // row 0 lives in vgpr0, lanes 0-15
  else        acc2[7] += bot;   // row 15 lives in vgpr7, lanes 16-31
  return acc2;
}

__global__ __launch_bounds__(WPB * 32) void guided_filter_wmma(
    const float* __restrict__ guide, const float* __restrict__ srcp,
    float* __restrict__ outp, int total_tiles) {
  __shared__ __align__(16) float lds[WPB * WAVE_FLOATS];
  const int wid = threadIdx.x >> 5;
  const int lane = threadIdx.x & 31;
  const int l = lane & 15;
  const int h = lane >> 4;
  float* lw = &lds[wid * WAVE_FLOATS];

  const int tile = blockIdx.x * WPB + wid;
  if (tile >= total_tiles) return;
  const int plane = tile / TILES_PER_PLANE;
  const int rem = tile - plane * TILES_PER_PLANE;
  const int ty = rem / TILES_X;
  const int tx = rem - ty * TILES_X;
  const int y0 = ty * 16, x0 = tx * 16;

  const float* gp = guide + (size_t)plane * (IMG_H * IMG_W);
  const float* sp = srcp + (size_t)plane * (IMG_H * IMG_W);
  float* op = outp + (size_t)plane * (IMG_H * IMG_W);

  // ---- load 20x20 guide/src tiles as 200 float-pairs each (b64/lane) ----
  // x0 is a multiple of 16 and pairs sit at even columns, so a pair is never
  // half in / half out of the image; OOB pairs are zero-filled (conv zero pad).
#pragma unroll
  for (int i = 0; i < 7; ++i) {
    const int p = lane + 32 * i;
    if (p < 200) {
      const int u = p / 10;
      const int vp = (p - u * 10) * 2;
      const int y = y0 - 2 + u;
      const int x = x0 - 2 + vp;
      const int li = u * 20 + vp;
      if ((unsigned)y < IMG_H && (unsigned)x < IMG_W) {
        const size_t goff = (size_t)y * IMG_W + x;
        async_b64(gp + goff, lw + G_OFF + li);
        async_b64(sp + goff, lw + S_OFF + li);
      } else {
        lw[G_OFF + li] = 0.f; lw[G_OFF + li + 1] = 0.f;
        lw[S_OFF + li] = 0.f; lw[S_OFF + li + 1] = 0.f;
      }
    }
  }
  wait_async0();

  // ---- stage 1a: horizontal 3-tap sums of g, s, g*s, g*g (20 rows x 18) ----
  for (int i = 0; i < 12; ++i) {
    const int p = lane + 32 * i;
    if (p < 360) {
      const int u = p / 18;
      const int c = p - u * 18;
      const float* gg = lw + G_OFF + u * 20 + c;
      const float* s2 = lw + S_OFF + u * 20 + c;
      const float g0 = gg[0], g1 = gg[1], g2 = gg[2];
      const float s0 = s2[0], s1 = s2[1], s22 = s2[2];
      const int hb = u * 18 + c;
      lw[HG_OFF + hb] = g0 + g1 + g2;
      lw[HS_OFF + hb] = s0 + s1 + s22;
      lw[HGS_OFF + hb] = g0 * s0 + g1 * s1 + g2 * s22;
      lw[HGG_OFF + hb] = g0 * g0 + g1 * g1 + g2 * g2;
    }
  }

  // ---- stage 1b: vertical sums, means, A and b at 18x18 ----
  for (int i = 0; i < 11; ++i) {
    const int p = lane + 32 * i;
    if (p < 324) {
      const int r = p / 18;
      const int c = p - r * 18;
      const int hb = r * 18 + c;
      const float sg = lw[HG_OFF + hb] + lw[HG_OFF + hb + 18] + lw[HG_OFF + hb + 36];
      const float ssum = lw[HS_OFF + hb] + lw[HS_OFF + hb + 18] + lw[HS_OFF + hb + 36];
      const float sgs = lw[HGS_OFF + hb] + lw[HGS_OFF + hb + 18] + lw[HGS_OFF + hb + 36];
      const float sgg = lw[HGG_OFF + hb] + lw[HGG_OFF + hb + 18] + lw[HGG_OFF + hb + 36];
      const int y = y0 - 1 + r, x = x0 - 1 + c;
      const bool inimg = ((unsigned)y < IMG_H) && ((unsigned)x < IMG_W);
      const float cy = 3.0f - (float)(y == 0) - (float)(y == IMG_H - 1);
      const float cx = 3.0f - (float)(x == 0) - (float)(x == IMG_W - 1);
      const float invN = 1.0f / (cy * cx);
      const float mg = sg * invN, ms = ssum * invN;
      const float mgs = sgs * invN, mgg = sgg * invN;
      const float A = (mgs - mg * ms) / (mgg - mg * mg + 0.01f);
      const float B = ms - A * mg;
      lw[AM_OFF + hb] = inimg ? A : 0.0f;  // zero outside image (zero padding)
      lw[BM_OFF + hb] = inimg ? B : 0.0f;
    }
  }

  // ---- stage 2: 3x3 box of A and b on the matrix pipe ----
  const v8f sa = box3x3_wmma(lw, AM_OFF, l, h);
  const v8f sb = box3x3_wmma(lw, BM_OFF, l, h);

  // ---- combine: out = (boxA/N)*guide + boxB/N ----
#pragma unroll
  for (int v = 0; v < 8; ++v) {
    const int py = 8 * h + v;  // C/D layout row
    const int y = y0 + py, x = x0 + l;
    const float cy = 3.0f - (float)(y == 0) - (float)(y == IMG_H - 1);
    const float cx = 3.0f - (float)(x == 0) - (float)(x == IMG_W - 1);
    const float invN = 1.0f / (cy * cx);
    const float gv = lw[G_OFF + (py + 2) * 20 + (l + 2)];
    op[(size_t)y * IMG_W + x] = (sa[v] * invN) * gv + sb[v] * invN;
  }
}

extern "C" void kernel_launch(void* const* d_in, const int* in_sizes, int n_in,
                              void* d_out, int out_size, void* d_ws,
                              size_t ws_size, hipStream_t stream) {
  (void)n_in; (void)out_size; (void)d_ws; (void)ws_size;
  const float* guide = (const float*)d_in[0];
  const float* src = (const float*)d_in[1];
  float* out = (float*)d_out;

  const int plane_elems = IMG_H * IMG_W;
  const int nplanes = in_sizes[0] / plane_elems;  // 16*3 = 48
  const int total_tiles = nplanes * TILES_PER_PLANE;
  const int blocks = (total_tiles + WPB - 1) / WPB;

  guided_filter_wmma<<<blocks, WPB * 32, 0, stream>>>(guide, src, out,
                                                      total_tiles);
}